// Seq2Seq_90787018703701
// MI455X (gfx1250) — compile-verified
//
#include <hip/hip_runtime.h>
#include <math.h>

#define HDIM 1024
#define SLEN 256
#define TLEN 256
#define VOUTN 32000
#define G4 (4 * HDIM) // 4096

typedef __attribute__((ext_vector_type(2))) float v2f;
typedef __attribute__((ext_vector_type(8))) float v8f;

__device__ __forceinline__ v2f ld2(const float* p) { return *(const v2f*)p; }

// ---------------------------------------------------------------------------
// init: zero the h double-buffer and the global barrier counter
// ---------------------------------------------------------------------------
__global__ void init_kernel(float* hbuf, unsigned* cnt) {
  int i = blockIdx.x * blockDim.x + threadIdx.x;
  if (i < 2 * HDIM) hbuf[i] = 0.0f;
  if (i == 0) *cnt = 0u;
}

// ---------------------------------------------------------------------------
// Pre-gate GEMM: pre[t][row] = emb[token[t]] . Wih[row] + bih[row] + bhh[row]
// M=256 (time), N=4096 (gate rows), K=1024.  One 16x16 tile per wave,
// f32 WMMA 16x16x4.  8192 tiles total (enc + dec).
// ---------------------------------------------------------------------------
__global__ __launch_bounds__(256) void pregate_gemm(
    const float* __restrict__ enc_emb, const float* __restrict__ dec_emb,
    const float* __restrict__ enc_Wih, const float* __restrict__ dec_Wih,
    const float* __restrict__ enc_bih, const float* __restrict__ enc_bhh,
    const float* __restrict__ dec_bih, const float* __restrict__ dec_bhh,
    const int* __restrict__ src, const int* __restrict__ trg,
    const int* __restrict__ start,
    float* __restrict__ encPre, float* __restrict__ decPre) {
  int wave = (blockIdx.x * blockDim.x + threadIdx.x) >> 5; // 0..8191
  int lane = threadIdx.x & 31;
  int isDec = wave >= 4096;
  int tile  = wave & 4095;
  int m0 = (tile >> 8) * 16;  // 16 M-tiles
  int n0 = (tile & 255) * 16; // 256 N-tiles
  int ml  = lane & 15;
  int hi  = lane >> 4; // K-half select
  int klo = hi * 2;

  const float* emb = isDec ? dec_emb : enc_emb;
  const float* Wih = isDec ? dec_Wih : enc_Wih;
  const float* bih = isDec ? dec_bih : enc_bih;
  const float* bhh = isDec ? dec_bhh : enc_bhh;
  float* pre = isDec ? decPre : encPre;

  // token gather for A-row (per M-lane)
  int m = m0 + ml;
  int tprev = (m > 0) ? (m - 1) : 0;
  int tok = isDec ? ((m == 0) ? start[0] : trg[tprev]) : src[m];
  const float* arow = emb + (long)tok * HDIM;
  const float* brow = Wih + (long)(n0 + ml) * HDIM; // B[k][n] = Wih[n][k]

  v8f acc = {0.f, 0.f, 0.f, 0.f, 0.f, 0.f, 0.f, 0.f};
#pragma unroll 4
  for (int k = 0; k < HDIM; k += 4) {
    v2f a = ld2(arow + k + klo);
    v2f b = ld2(brow + k + klo);
    acc = __builtin_amdgcn_wmma_f32_16x16x4_f32(false, a, false, b, (short)0,
                                                acc, false, false);
  }

  int n = n0 + ml;
  float bias = bih[n] + bhh[n];
#pragma unroll
  for (int r = 0; r < 8; ++r) {
    int mm = m0 + r + 8 * hi; // C layout: lanes>=16 hold M=m0+8..m0+15
    pre[(long)mm * G4 + n] = acc[r] + bias;
  }
}

// ---------------------------------------------------------------------------
// Persistent LSTM recurrence: 128 WGs x 256 threads, one global spin barrier
// per step.  WG w owns j in [8w, 8w+8).  Its 32 Whh rows (4 gates x 8 j,
// 128 KB) are pinned in LDS (loaded once per phase: enc at t=0, dec at
// t=SLEN), so each step touches only 4 KB of global (the h vector).
// c lives in registers of threads 0..7.  Decoder h_t rows go to Hs.
// ---------------------------------------------------------------------------
#define NWG 128
#define RTHREADS 256
#define JSL 8 // j per WG

__global__ __launch_bounds__(RTHREADS) void lstm_recurrence(
    const float* __restrict__ encPre, const float* __restrict__ decPre,
    const float* __restrict__ enc_Whh, const float* __restrict__ dec_Whh,
    float* hbuf,            // 2*HDIM double buffer
    float* __restrict__ Hs, // TLEN*HDIM decoder hidden states
    unsigned* cnt) {
  __shared__ __align__(16) float wsh[32 * HDIM]; // 128 KB weight slice
  __shared__ __align__(16) float hsh[HDIM];
  __shared__ float partial[RTHREADS];
  __shared__ float gsh[32]; // [gate*8 + j]

  int tid = threadIdx.x;
  int jbase = blockIdx.x * JSL;
  int p = tid >> 3; // (gate,j) pair 0..31 : gate = p>>3, j = p&7
  int q = tid & 7;  // k-slice 0..7
  int k0 = q * 128;

  float cstate = 0.0f; // valid for tid < 8 (j = jbase + tid)

  for (int t = 0; t < SLEN + TLEN; ++t) {
    bool dec = (t >= SLEN);
    const float* pre = dec ? (decPre + (long)(t - SLEN) * G4)
                           : (encPre + (long)t * G4);
    const volatile float* hrd = hbuf + (t & 1) * HDIM;
    float* hwr = hbuf + ((t + 1) & 1) * HDIM;

    // (re)load the 32-row Whh slice into LDS at each phase start
    if (t == 0 || t == SLEN) {
      const float* W = dec ? dec_Whh : enc_Whh;
#pragma unroll 4
      for (int w = 0; w < 32; ++w) { // row w: gate = w>>3, j = jbase + (w&7)
        long grow = (long)((w >> 3) * HDIM + jbase + (w & 7));
        float4 v = *(const float4*)(W + grow * HDIM + tid * 4);
        *(float4*)(wsh + w * HDIM + tid * 4) = v;
      }
      __syncthreads();
    }

    // stage h into LDS (volatile: always refetch cross-WGP data)
    hsh[tid] = hrd[tid];
    hsh[tid + 256] = hrd[tid + 256];
    hsh[tid + 512] = hrd[tid + 512];
    hsh[tid + 768] = hrd[tid + 768];
    __syncthreads();

    // 128-element partial dot product from LDS, float4 vectorized
    const float4* wr = (const float4*)(wsh + p * HDIM + k0);
    const float4* h4 = (const float4*)(hsh + k0);
    float s = 0.0f;
#pragma unroll 8
    for (int kk = 0; kk < 32; ++kk) {
      float4 w = wr[kk];
      float4 h = h4[kk];
      s += w.x * h.x + w.y * h.y + w.z * h.z + w.w * h.w;
    }
    partial[tid] = s;
    __syncthreads();

    if (tid < 32) {
      float g = 0.0f;
#pragma unroll
      for (int u = 0; u < 8; ++u) g += partial[tid * 8 + u];
      int rrow = (tid >> 3) * HDIM + jbase + (tid & 7);
      gsh[tid] = g + pre[rrow];
    }
    __syncthreads();

    if (tid < 8) {
      float iv = gsh[tid];
      float fv = gsh[8 + tid];
      float gv = gsh[16 + tid];
      float ov = gsh[24 + tid];
      iv = 1.0f / (1.0f + __expf(-iv));
      fv = 1.0f / (1.0f + __expf(-fv));
      gv = tanhf(gv);
      ov = 1.0f / (1.0f + __expf(-ov));
      cstate = fv * cstate + iv * gv;
      float hn = ov * tanhf(cstate);
      hwr[jbase + tid] = hn;
      if (dec) Hs[(long)(t - SLEN) * HDIM + jbase + tid] = hn;
    }

    // device-scope release of h writes, then global barrier (monotonic count)
    __threadfence();
    __syncthreads();
    if (tid == 0) {
      atomicAdd(cnt, 1u);
      unsigned target = (unsigned)(t + 1) * NWG;
      while (*(volatile unsigned*)cnt < target) {
        __builtin_amdgcn_s_sleep(1);
      }
    }
    __syncthreads();
    __threadfence(); // acquire before reading next step's h
  }
}

// ---------------------------------------------------------------------------
// Output GEMM: logits[256][32000] = Hs @ lin_W.T + lin_b  (f32 WMMA).
// Each wave computes a 64x16 strip (4 M-tiles share each B load).
// ---------------------------------------------------------------------------
__global__ __launch_bounds__(256) void out_gemm(
    const float* __restrict__ Hs, const float* __restrict__ lin_W,
    const float* __restrict__ lin_b, float* __restrict__ out) {
  int wave = (blockIdx.x * blockDim.x + threadIdx.x) >> 5; // 0..7999
  int lane = threadIdx.x & 31;
  int nt = wave % 2000;
  int ms = wave / 2000; // 0..3
  int n0 = nt * 16;
  int m0 = ms * 64;
  int ml = lane & 15;
  int hi = lane >> 4;
  int klo = hi * 2;

  const float* brow = lin_W + (long)(n0 + ml) * HDIM;
  const float* a0 = Hs + (long)(m0 + ml) * HDIM;
  const float* a1 = a0 + 16 * HDIM;
  const float* a2 = a0 + 32 * HDIM;
  const float* a3 = a0 + 48 * HDIM;

  v8f c0 = {0.f, 0.f, 0.f, 0.f, 0.f, 0.f, 0.f, 0.f};
  v8f c1 = c0, c2 = c0, c3 = c0;

#pragma unroll 4
  for (int k = 0; k < HDIM; k += 4) {
    v2f b  = ld2(brow + k + klo);
    v2f x0 = ld2(a0 + k + klo);
    v2f x1 = ld2(a1 + k + klo);
    v2f x2 = ld2(a2 + k + klo);
    v2f x3 = ld2(a3 + k + klo);
    c0 = __builtin_amdgcn_wmma_f32_16x16x4_f32(false, x0, false, b, (short)0, c0, false, false);
    c1 = __builtin_amdgcn_wmma_f32_16x16x4_f32(false, x1, false, b, (short)0, c1, false, false);
    c2 = __builtin_amdgcn_wmma_f32_16x16x4_f32(false, x2, false, b, (short)0, c2, false, false);
    c3 = __builtin_amdgcn_wmma_f32_16x16x4_f32(false, x3, false, b, (short)0, c3, false, false);
  }

  int n = n0 + ml;
  float bias = lin_b[n];
#pragma unroll
  for (int r = 0; r < 8; ++r) {
    int mm = r + 8 * hi;
    out[(long)(m0 + mm) * VOUTN + n]      = c0[r] + bias;
    out[(long)(m0 + 16 + mm) * VOUTN + n] = c1[r] + bias;
    out[(long)(m0 + 32 + mm) * VOUTN + n] = c2[r] + bias;
    out[(long)(m0 + 48 + mm) * VOUTN + n] = c3[r] + bias;
  }
}

// ---------------------------------------------------------------------------
// In-place log-softmax, one workgroup per row of 32000 logits.
// ---------------------------------------------------------------------------
__global__ __launch_bounds__(1024) void logsoftmax_kernel(float* __restrict__ out) {
  __shared__ float red[1024];
  int tid = threadIdx.x;
  float* x = out + (long)blockIdx.x * VOUTN;

  float mx = -INFINITY;
  for (int i = tid; i < VOUTN; i += 1024) mx = fmaxf(mx, x[i]);
  red[tid] = mx;
  __syncthreads();
  for (int s = 512; s > 0; s >>= 1) {
    if (tid < s) red[tid] = fmaxf(red[tid], red[tid + s]);
    __syncthreads();
  }
  mx = red[0];
  __syncthreads();

  float sum = 0.0f;
  for (int i = tid; i < VOUTN; i += 1024) sum += expf(x[i] - mx);
  red[tid] = sum;
  __syncthreads();
  for (int s = 512; s > 0; s >>= 1) {
    if (tid < s) red[tid] += red[tid + s];
    __syncthreads();
  }
  float lse = mx + logf(red[0]);
  __syncthreads();

  for (int i = tid; i < VOUTN; i += 1024) x[i] = x[i] - lse;
}

// ---------------------------------------------------------------------------
// launch
// ---------------------------------------------------------------------------
extern "C" void kernel_launch(void* const* d_in, const int* in_sizes, int n_in,
                              void* d_out, int out_size, void* d_ws,
                              size_t ws_size, hipStream_t stream) {
  (void)in_sizes; (void)n_in; (void)out_size; (void)ws_size;

  const int* src    = (const int*)d_in[0];
  const int* trg    = (const int*)d_in[1];
  const int* start  = (const int*)d_in[2];
  const float* enc_emb = (const float*)d_in[3];
  const float* enc_Wih = (const float*)d_in[4];
  const float* enc_Whh = (const float*)d_in[5];
  const float* enc_bih = (const float*)d_in[6];
  const float* enc_bhh = (const float*)d_in[7];
  const float* dec_emb = (const float*)d_in[8];
  const float* dec_Wih = (const float*)d_in[9];
  const float* dec_Whh = (const float*)d_in[10];
  const float* dec_bih = (const float*)d_in[11];
  const float* dec_bhh = (const float*)d_in[12];
  const float* lin_W   = (const float*)d_in[13];
  const float* lin_b   = (const float*)d_in[14];
  float* out = (float*)d_out;

  float* ws = (float*)d_ws;
  float* encPre = ws;                        // 256*4096
  float* decPre = ws + 1048576;              // 256*4096
  float* Hs     = ws + 2097152;              // 256*1024
  float* hbuf   = ws + 2359296;              // 2*1024
  unsigned* cnt = (unsigned*)(ws + 2361344); // 1

  init_kernel<<<8, 256, 0, stream>>>(hbuf, cnt);
  pregate_gemm<<<1024, 256, 0, stream>>>(enc_emb, dec_emb, enc_Wih, dec_Wih,
                                         enc_bih, enc_bhh, dec_bih, dec_bhh,
                                         src, trg, start, encPre, decPre);
  lstm_recurrence<<<NWG, RTHREADS, 0, stream>>>(encPre, decPre, enc_Whh,
                                                dec_Whh, hbuf, Hs, cnt);
  out_gemm<<<1000, 256, 0, stream>>>(Hs, lin_W, lin_b, out);
  logsoftmax_kernel<<<256, 1024, 0, stream>>>(out);
}